// GraphConvolution_69372311765224
// MI455X (gfx1250) — compile-verified
//
#include <hip/hip_runtime.h>

typedef __attribute__((ext_vector_type(2))) float v2f;
typedef __attribute__((ext_vector_type(4))) float v4f;
typedef __attribute__((ext_vector_type(8))) float v8f;
typedef __attribute__((ext_vector_type(4))) int   v4i;

#define AS_GLOBAL __attribute__((address_space(1)))
#define AS_LDS    __attribute__((address_space(3)))

#if __has_builtin(__builtin_amdgcn_global_load_async_to_lds_b32) && \
    __has_builtin(__builtin_amdgcn_global_load_async_to_lds_b128)
#define HAVE_ASYNC_LDS 1
#else
#define HAVE_ASYNC_LDS 0
#endif

__device__ __forceinline__ void copy_g2l_b32(const float* g, float* l)
{
#if HAVE_ASYNC_LDS
    __builtin_amdgcn_global_load_async_to_lds_b32(
        (AS_GLOBAL int*)g, (AS_LDS int*)l, 0, 0);
#else
    *l = *g;
#endif
}

__device__ __forceinline__ void copy_g2l_b128(const float* g, float* l)
{
#if HAVE_ASYNC_LDS
    __builtin_amdgcn_global_load_async_to_lds_b128(
        (AS_GLOBAL v4i*)g, (AS_LDS v4i*)l, 0, 0);
#else
    *(v4f*)l = *(const v4f*)g;
#endif
}

__device__ __forceinline__ void wait_async_all()
{
#if HAVE_ASYNC_LDS
#if __has_builtin(__builtin_amdgcn_s_wait_asynccnt)
    __builtin_amdgcn_s_wait_asynccnt(0);
#else
    asm volatile("s_wait_asynccnt 0" ::: "memory");
#endif
#endif
}

// C(MxN) = A(MxK) @ B(KxN), fp32, via V_WMMA_F32_16X16X4_F32, wave32.
// TA: A[m,k] = Ap[k*lda + m]   (transposed storage)
// TB: B[k,n] = Bp[n*ldb + k]   (transposed storage)
// Block tile 128x128, BK=16, 256 threads = 8 waves (4x2 wave grid),
// each wave owns 32(m) x 64(n) = 2x4 accumulators of 16x16.
// LDS: A tile [m][k] pitch 20, B tile [n][k] pitch 20 -> both fragments are
// single ds_load_b64; stride-20 rows give conflict-free banks for 16-lane
// groups; upper half-wave reads k+2 (disjoint bank sets).
// Global->LDS traffic is double-buffered via GLOBAL_LOAD_ASYNC_TO_LDS (ASYNCcnt).
template <bool TA, bool TB>
__global__ __launch_bounds__(256)
void gemm_wmma_f32(const float* __restrict__ A, const float* __restrict__ B,
                   float* __restrict__ C, int K, int lda, int ldb, int ldc)
{
    constexpr int BM = 128, BN = 128, BK = 16;
    constexpr int APAD = 20, BPITCH = 20;

    __shared__ float As[2][BM * APAD];   // 2 x 2560 floats
    __shared__ float Bs[2][BN * BPITCH]; // 2 x 2560 floats  (total LDS = 40 KB)

    const int tid  = threadIdx.x;
    const int wid  = tid >> 5;
    const int lane = tid & 31;
    const int half = lane >> 4;
    const int l15  = lane & 15;
    const int wm   = (wid & 3) * 32;
    const int wn   = (wid >> 2) * 64;
    const int bm   = blockIdx.y * BM;
    const int bn   = blockIdx.x * BN;

    auto load_tile = [&](int buf, int kt) {
        // ---- A tile -> As[m][k] ----
        if (!TA) {
            // row-major A: 16 contiguous floats per tile row -> async b128
#pragma unroll
            for (int j = 0; j < 2; ++j) {
                const int idx = tid * 4 + j * 1024;
                const int r = idx >> 4, c = idx & 15;
                copy_g2l_b128(A + (size_t)(bm + r) * lda + (kt + c),
                              &As[buf][r * APAD + c]);
            }
        } else {
            // A[m,k]=Ap[k*lda+m]: contiguous in m across lanes -> coalesced async b32 scatter
#pragma unroll
            for (int j = 0; j < 8; ++j) {
                const int idx = j * 256 + tid;
                const int kf = idx >> 7, m = idx & 127;
                copy_g2l_b32(A + (size_t)(kt + kf) * lda + (bm + m),
                             &As[buf][m * APAD + kf]);
            }
        }
        // ---- B tile -> Bs[n][k] ----
        if (TB) {
            // B[k,n]=Bp[n*ldb+k]: 16 contiguous floats per n -> async b128, contiguous both sides
#pragma unroll
            for (int j = 0; j < 2; ++j) {
                const int idx = tid * 4 + j * 1024;
                const int n = idx >> 4, c = idx & 15;
                copy_g2l_b128(B + (size_t)(bn + n) * ldb + (kt + c),
                              &Bs[buf][n * BPITCH + c]);
            }
        } else {
            // row-major B: contiguous in n across lanes -> coalesced async b32 scatter into [n][k]
#pragma unroll
            for (int j = 0; j < 8; ++j) {
                const int idx = j * 256 + tid;
                const int kf = idx >> 7, n = idx & 127;
                copy_g2l_b32(B + (size_t)(kt + kf) * ldb + (bn + n),
                             &Bs[buf][n * BPITCH + kf]);
            }
        }
    };

    v8f acc[2][4];
#pragma unroll
    for (int mf = 0; mf < 2; ++mf)
#pragma unroll
        for (int nf = 0; nf < 4; ++nf)
#pragma unroll
            for (int i = 0; i < 8; ++i) acc[mf][nf][i] = 0.0f;

    load_tile(0, 0);
    int buf = 0;
    for (int kt = 0; kt < K; kt += BK) {
        wait_async_all();     // my tile-(kt) async loads complete
        __syncthreads();      // everyone's complete; everyone done reading other buffer

        if (kt + BK < K) load_tile(buf ^ 1, kt + BK);  // prefetch overlaps compute

#pragma unroll
        for (int k0 = 0; k0 < BK; k0 += 4) {
            const int kk = k0 + half * 2;  // lanes 16..31 carry K=+2,+3 (f32 WMMA layout)
            v2f afr[2], bfr[4];
#pragma unroll
            for (int mf = 0; mf < 2; ++mf)
                afr[mf] = *(const v2f*)(&As[buf][(wm + mf * 16 + l15) * APAD + kk]);
#pragma unroll
            for (int nf = 0; nf < 4; ++nf)
                bfr[nf] = *(const v2f*)(&Bs[buf][(wn + nf * 16 + l15) * BPITCH + kk]);
#pragma unroll
            for (int mf = 0; mf < 2; ++mf)
#pragma unroll
                for (int nf = 0; nf < 4; ++nf)
                    acc[mf][nf] = __builtin_amdgcn_wmma_f32_16x16x4_f32(
                        false, afr[mf], false, bfr[nf],
                        (short)0, acc[mf][nf], false, false);
        }
        buf ^= 1;
    }

    // ---- store (nontemporal: keep 192MB L2 for the adj/S^T streams) ----
#pragma unroll
    for (int mf = 0; mf < 2; ++mf)
#pragma unroll
        for (int nf = 0; nf < 4; ++nf) {
            const int col = bn + wn + nf * 16 + l15;
#pragma unroll
            for (int i = 0; i < 8; ++i) {
                const int row = bm + wm + mf * 16 + half * 8 + i;
                __builtin_nontemporal_store(acc[mf][nf][i], &C[(size_t)row * ldc + col]);
            }
        }
}

__global__ void copy_f4(const float* __restrict__ src, float* __restrict__ dst, int n4)
{
    const int i = blockIdx.x * blockDim.x + threadIdx.x;
    if (i < n4) {
        const v4f v = ((const v4f*)src)[i];
        __builtin_nontemporal_store(v, &((v4f*)dst)[i]);
    }
}

extern "C" void kernel_launch(void* const* d_in, const int* in_sizes, int n_in,
                              void* d_out, int out_size, void* d_ws, size_t ws_size,
                              hipStream_t stream)
{
    const int NN = 8192;              // num_nodes == out_features
    const int F  = in_sizes[0] / NN;  // in_features = 512

    const float* X   = (const float*)d_in[0];  // (NN, F)
    const float* adj = (const float*)d_in[1];  // (NN, NN)
    const float* W   = (const float*)d_in[2];  // (F, NN)
    float* out = (float*)d_out;                // (NN, NN) output, then (F, NN) weight
    float* ST  = (float*)d_ws;                 // scratch: support^T (NN, NN) fp32

    dim3 grid(NN / 128, NN / 128), block(256);

    // GEMM1: ST = W^T (NNxF) @ X^T (FxNN)  == support.T
    gemm_wmma_f32<true, true><<<grid, block, 0, stream>>>(W, X, ST, F, NN, F, NN);

    // GEMM2: out = adj (NNxNN) @ ST (NNxNN)
    gemm_wmma_f32<false, false><<<grid, block, 0, stream>>>(adj, ST, out, NN, NN, NN, NN);

    // Tuple tail: weight after the (NN*NN) output block.
    const int n4 = (F * NN) / 4;
    copy_f4<<<(n4 + 255) / 256, 256, 0, stream>>>(W, out + (size_t)NN * NN, n4);
}